// AdaptivePooling_87840671137912
// MI455X (gfx1250) — compile-verified
//
#include <hip/hip_runtime.h>
#include <hip/hip_bf16.h>

// Adaptive max pool [32,224,224,128] f32 -> [32,7,7,128] f32.
// 224 / 7 = 32 exactly -> uniform 32x32 bins.
//
// Pure HBM-streaming reduction (822 MB read once, ~0.8 MB written).
// Optimal path on MI455X: coalesced global_load_b128 with NT temporal hint
// (input is 4x the 192MB L2 and single-use -> don't thrash L2), running max
// in VGPRs, tiny LDS cross-wave reduction. No matmul structure -> no WMMA.

typedef float v4f __attribute__((ext_vector_type(4)));

#define BINS      7
#define BIN_SZ    32          // 224 / 7
#define CH        128
#define WIDTH     224
#define HEIGHT    224
#define ROWS_PER_WAVE 4       // 32 rows / 8 waves

static __device__ __forceinline__ v4f v4max(v4f a, v4f b) {
    v4f r;
    r.x = __builtin_fmaxf(a.x, b.x);
    r.y = __builtin_fmaxf(a.y, b.y);
    r.z = __builtin_fmaxf(a.z, b.z);
    r.w = __builtin_fmaxf(a.w, b.w);
    return r;
}

__global__ __launch_bounds__(256, 4)
void adaptive_maxpool_f32_kernel(const float* __restrict__ in,
                                 float* __restrict__ out) {
    const int lane  = threadIdx.x & 31;   // channel quad: channels 4*lane..4*lane+3
    const int wave  = threadIdx.x >> 5;   // 0..7 : row-slice of the 32-row bin

    const int wb = blockIdx.x % BINS;
    const int hb = (blockIdx.x / BINS) % BINS;
    const int b  = blockIdx.x / (BINS * BINS);

    const long rowStride = (long)WIDTH * CH;                 // floats per input row
    const int  row0      = hb * BIN_SZ + wave * ROWS_PER_WAVE;
    const long base      = ((long)b * HEIGHT + row0) * rowStride
                         + (long)(wb * BIN_SZ) * CH
                         + (long)lane * 4;

    v4f m;
    m.x = m.y = m.z = m.w = -__builtin_inff();

    // 4 rows x 32 cols of B128 NT loads per thread; consecutive lanes cover
    // one pixel's 512B channel line (perfect coalescing), consecutive cols are
    // contiguous 512B chunks -> pure sequential streaming.
    for (int r = 0; r < ROWS_PER_WAVE; ++r) {
        const float* p = in + base + (long)r * rowStride;
        #pragma unroll
        for (int col = 0; col < BIN_SZ; ++col) {
            // col*CH floats = col*512 bytes: fits the 24-bit immediate offset,
            // so these become one dense clause of global_load_b128 ... nt.
            v4f v = __builtin_nontemporal_load((const v4f*)(p + (long)col * CH));
            m = v4max(m, v);
        }
    }

    // Cross-wave reduction: 8 partials per channel-quad through LDS (4 KB).
    __shared__ v4f red[8][32];
    red[wave][lane] = m;
    __syncthreads();

    if (wave == 0) {
        v4f acc = red[0][lane];
        #pragma unroll
        for (int s = 1; s < 8; ++s) acc = v4max(acc, red[s][lane]);

        const long o = (((long)b * BINS + hb) * BINS + wb) * CH + (long)lane * 4;
        *(v4f*)(out + o) = acc;   // global_store_b128, coalesced
    }
}

extern "C" void kernel_launch(void* const* d_in, const int* in_sizes, int n_in,
                              void* d_out, int out_size, void* d_ws, size_t ws_size,
                              hipStream_t stream) {
    (void)in_sizes; (void)n_in; (void)out_size; (void)d_ws; (void)ws_size;
    const float* in  = (const float*)d_in[0];
    float*       out = (float*)d_out;

    const int B = 32;
    dim3 grid(B * BINS * BINS);   // 1568 workgroups, one per output pixel
    dim3 block(256);              // 8 wave32s
    adaptive_maxpool_f32_kernel<<<grid, block, 0, stream>>>(in, out);
}